// DbrxRouter_65816078844559
// MI455X (gfx1250) — compile-verified
//
#include <hip/hip_runtime.h>

typedef __attribute__((ext_vector_type(2))) float v2f;
typedef __attribute__((ext_vector_type(8))) float v8f;

#define DIM   6144
#define NEXP  16
#define TOPK  4
#define NTOK  (4 * 4096)

// DPP16 controls (LLVM encodings)
#define DPP_XOR1  0xB1    // quad_perm [1,0,3,2]
#define DPP_XOR2  0x4E    // quad_perm [2,3,0,1]
#define DPP_XOR7  0x141   // ROW_HALF_MIRROR
#define DPP_XOR15 0x140   // ROW_MIRROR

template <int CTRL>
__device__ __forceinline__ int dpp_i(int v) {
    return __builtin_amdgcn_update_dpp(0, v, CTRL, 0xF, 0xF, true);
}
template <int CTRL>
__device__ __forceinline__ float dpp_f(float v) {
    return __int_as_float(dpp_i<CTRL>(__float_as_int(v)));
}

// 16-lane butterfly max / sum (pure VALU via DPP16)
__device__ __forceinline__ float grp16_max(float v) {
    v = fmaxf(v, dpp_f<DPP_XOR1>(v));
    v = fmaxf(v, dpp_f<DPP_XOR2>(v));
    v = fmaxf(v, dpp_f<DPP_XOR7>(v));
    v = fmaxf(v, dpp_f<DPP_XOR15>(v));
    return v;
}
__device__ __forceinline__ float grp16_sum(float v) {
    v += dpp_f<DPP_XOR1>(v);
    v += dpp_f<DPP_XOR2>(v);
    v += dpp_f<DPP_XOR7>(v);
    v += dpp_f<DPP_XOR15>(v);
    return v;
}

// one arg-max butterfly step: prefer larger value, tie -> lower index
template <int CTRL>
__device__ __forceinline__ void argmax_step(float& bv, int& bi) {
    float ov = dpp_f<CTRL>(bv);
    int   oi = dpp_i<CTRL>(bi);
    if (ov > bv || (ov == bv && oi < bi)) { bv = ov; bi = oi; }
}

__global__ __launch_bounds__(256) void
dbrx_router_wmma(const float* __restrict__ x,
                 const float* __restrict__ W,
                 float* __restrict__ weights_out,   // [NTOK, 16]
                 float* __restrict__ topw_out,      // [NTOK, 4]
                 int*   __restrict__ tope_out)      // [NTOK, 4]
{
    const int lane  = threadIdx.x & 31;
    const int wave  = threadIdx.x >> 5;
    const int tile  = blockIdx.x * 8 + wave;     // 16 tokens per wave-tile
    const int tok0  = tile * 16;
    const int laneM = lane & 15;                 // row (A) / expert col (B,C)
    const int hi    = lane >> 4;                 // K-pair selector

    // A: lane holds x[tok0+laneM][kb + 2*hi + {0,1}]
    const float* aPtr = x + (size_t)(tok0 + laneM) * DIM + 2 * hi;
    // B: lane holds W[(kb + 2*hi + {0,1})*16 + laneM]
    const float* bPtr = W + 2 * hi * NEXP + laneM;

    v8f c = {};
#pragma unroll 8
    for (int kb = 0; kb < DIM; kb += 4) {
        v2f a = *(const v2f*)(aPtr + kb);
        v2f b;
        b.x = bPtr[kb * NEXP];
        b.y = bPtr[kb * NEXP + NEXP];
        // D = A(16x4) * B(4x16) + C  -- v_wmma_f32_16x16x4_f32
        c = __builtin_amdgcn_wmma_f32_16x16x4_f32(
                /*neg_a=*/false, a, /*neg_b=*/false, b,
                /*c_mod=*/(short)0, c, /*reuse_a=*/false, /*reuse_b=*/false);
    }

    const float NEG_INF = -__builtin_inff();

    // c[r]: lanes 0-15  -> token tok0+r,   expert = laneM
    //       lanes 16-31 -> token tok0+8+r, expert = laneM
#pragma unroll
    for (int r = 0; r < 8; ++r) {
        const int token = tok0 + r + 8 * hi;
        const float logit = c[r];

        // ---- softmax over the 16 lanes of this half-wave group ----
        float m  = grp16_max(logit);
        float ex = __expf(logit - m);
        float s  = grp16_sum(ex);
        float prob = ex / s;
        weights_out[token * NEXP + laneM] = prob;

        // ---- top-4 (ties -> lower index, matching jax.lax.top_k) ----
        float cur = prob;
        float tw0, tw1, tw2, tw3;
        int   te0, te1, te2, te3;
#pragma unroll
        for (int t = 0; t < TOPK; ++t) {
            float bv = cur;
            int   bi = laneM;
            argmax_step<DPP_XOR1>(bv, bi);
            argmax_step<DPP_XOR2>(bv, bi);
            argmax_step<DPP_XOR7>(bv, bi);
            argmax_step<DPP_XOR15>(bv, bi);
            if (t == 0)      { tw0 = bv; te0 = bi; }
            else if (t == 1) { tw1 = bv; te1 = bi; }
            else if (t == 2) { tw2 = bv; te2 = bi; }
            else             { tw3 = bv; te3 = bi; }
            if (laneM == bi) cur = NEG_INF;   // knock out winner
        }
        const float inv = 1.0f / (tw0 + tw1 + tw2 + tw3);
        if (laneM < TOPK) {
            float twv = (laneM == 0) ? tw0 : (laneM == 1) ? tw1
                        : (laneM == 2) ? tw2 : tw3;
            int   tev = (laneM == 0) ? te0 : (laneM == 1) ? te1
                        : (laneM == 2) ? te2 : te3;
            topw_out[token * TOPK + laneM] = twv * inv;
            tope_out[token * TOPK + laneM] = tev;
        }
    }
}

extern "C" void kernel_launch(void* const* d_in, const int* in_sizes, int n_in,
                              void* d_out, int out_size, void* d_ws, size_t ws_size,
                              hipStream_t stream)
{
    (void)in_sizes; (void)n_in; (void)d_ws; (void)ws_size; (void)out_size;
    const float* x = (const float*)d_in[0];
    const float* W = (const float*)d_in[1];

    float* out     = (float*)d_out;
    float* weights = out;                              // NTOK*16
    float* topw    = out + (size_t)NTOK * NEXP;        // NTOK*4
    int*   tope    = (int*)(out + (size_t)NTOK * NEXP + (size_t)NTOK * TOPK);

    // 16384 tokens / 16 per wave / 8 waves per block = 128 blocks
    dbrx_router_wmma<<<NTOK / 16 / 8, 256, 0, stream>>>(x, W, weights, topw, tope);
}